// AUCSurrogateLoss_22136261443962
// MI455X (gfx1250) — compile-verified
//
#include <hip/hip_runtime.h>
#include <hip/hip_bf16.h>

// AUC surrogate loss, B x B pairwise (B=16384 -> 268M pairs, compute-bound).
//
// Key trick: diff tile is rank-2, so one V_WMMA_F32_16X16X4_F32 produces a
// 16x16 tile of (q_i - q_j) with q = y_pred*log2(e); VALU then only does
// v_exp_f32 + add + v_rcp_f32 per element. Class masks are folded in as
// +3e38 sentinels (exp->inf, rcp->0), so no divergence and EXEC stays all-1s
// as WMMA requires. All reductions fixed-order => deterministic output.

typedef __attribute__((ext_vector_type(2))) float v2f;
typedef __attribute__((ext_vector_type(8))) float v8f;

#define LOG2E 1.4426950408889634f
#define SENTINEL 3.0e38f
#define STRIP 512           // j-values staged in LDS per block
#define WAVES_PER_BLOCK 8   // 256 threads, wave32
#define ITILE (WAVES_PER_BLOCK * 16)  // 128 i-rows per block

// ---------------------------------------------------------------------------
// Prep: q = y_pred * log2(e); masked entries -> +SENTINEL so their pairs
// contribute exactly 0 after exp/rcp. Pads [n, padB) with sentinels too.
// ---------------------------------------------------------------------------
__global__ void __launch_bounds__(256)
auc_prep_kernel(const float* __restrict__ y_pred,
                const int* __restrict__ y_true,
                float* __restrict__ qpos,   // +q if label==1 else SENTINEL
                float* __restrict__ qneg,   // -q if label==0 else SENTINEL
                int n, int padB) {
    int i = blockIdx.x * 256 + threadIdx.x;
    if (i >= padB) return;
    bool valid = (i < n);
    float p = valid ? y_pred[i] : 0.0f;
    int   t = valid ? y_true[i] : -1;
    float q = p * LOG2E;
    qpos[i] = (t == 1) ?  q : SENTINEL;
    qneg[i] = (t == 0) ? -q : SENTINEL;
}

// ---------------------------------------------------------------------------
// Main pairwise kernel. Each wave owns a 16-row i-tile; block stages a 512
// element j-strip into LDS; wave sweeps 32 WMMA tiles per strip.
//   D = A x B, A(16x4) = [qpos_i, 1, 0, 0], B(4x16) = [1; qneg_j; 0; 0]
//   => D[i][j] = qpos[i] + qneg[j] = (p_i - p_j)*log2(e) for live pairs.
//   sigma(-(p_i-p_j)) = 1 / (1 + 2^D)   -> v_exp_f32 + v_rcp_f32.
// ---------------------------------------------------------------------------
__global__ void __launch_bounds__(256)
auc_pair_kernel(const float* __restrict__ qpos,
                const float* __restrict__ qneg,
                float* __restrict__ partials) {
    __shared__ float lds_qn[STRIP];
    __shared__ float wave_sum[WAVES_PER_BLOCK];

    const int lane  = threadIdx.x & 31;
    const int wave  = threadIdx.x >> 5;
    const int half  = lane & 15;          // M (for A) or N (for B) index
    const bool lo16 = (lane < 16);        // lanes 0..15 hold K=0,1; 16..31 hold K=2,3 (zero)

    // Stage this block's j-strip into LDS (coalesced b32 loads).
    const int j0 = blockIdx.y * STRIP;
    for (int t = threadIdx.x; t < STRIP; t += 256)
        lds_qn[t] = qneg[j0 + t];
    __syncthreads();

    // A operand: lanes 0..15 -> VGPR0 = K0 = qpos[i], VGPR1 = K1 = 1.0
    //            lanes 16..31 -> K2 = K3 = 0
    const int iBase = (blockIdx.x * WAVES_PER_BLOCK + wave) * 16;
    float qa = qpos[iBase + half];
    v2f A;
    A.x = lo16 ? qa   : 0.0f;
    A.y = lo16 ? 1.0f : 0.0f;

    float acc = 0.0f;
    for (int jt = 0; jt < STRIP; jt += 16) {
        // B operand: lanes 0..15 -> VGPR0 = K0-row = 1.0, VGPR1 = K1-row = qneg[j]
        float qb = lds_qn[jt + half];
        v2f Bm;
        Bm.x = lo16 ? 1.0f : 0.0f;
        Bm.y = lo16 ? qb   : 0.0f;

        v8f C = {};
        // 8 args: (neg_a, A, neg_b, B, c_mod, C, reuse_a, reuse_b)
        v8f D = __builtin_amdgcn_wmma_f32_16x16x4_f32(
            false, A, false, Bm, (short)0, C, false, false);

        // Nonlinearity on the 8 accumulator VGPRs: s = 1/(1 + 2^D).
        // Sentinel pairs: D huge -> exp=+inf -> rcp(inf)=0 (exact mask).
#pragma unroll
        for (int k = 0; k < 8; ++k) {
            float e = __builtin_amdgcn_exp2f(D[k]);      // v_exp_f32
            acc += __builtin_amdgcn_rcpf(1.0f + e);      // v_rcp_f32
        }
    }

    // Fixed-order wave32 reduction.
#pragma unroll
    for (int off = 16; off >= 1; off >>= 1)
        acc += __shfl_xor(acc, off, 32);
    if (lane == 0) wave_sum[wave] = acc;
    __syncthreads();

    if (threadIdx.x == 0) {
        float s = 0.0f;
#pragma unroll
        for (int w = 0; w < WAVES_PER_BLOCK; ++w) s += wave_sum[w];
        partials[blockIdx.y * gridDim.x + blockIdx.x] = s;  // deterministic slot
    }
}

// ---------------------------------------------------------------------------
// Finalize: deterministic tree-sum of per-block partials + exact integer
// class count; loss = numerator / (n_pos * n_neg).
// ---------------------------------------------------------------------------
__global__ void __launch_bounds__(256)
auc_finalize_kernel(const int* __restrict__ y_true,
                    const float* __restrict__ partials,
                    int nPart, int n, float* __restrict__ out) {
    __shared__ float fs[256];
    __shared__ int   cs[256];
    float s = 0.0f;
    int   c = 0;
    for (int t = threadIdx.x; t < nPart; t += 256) s += partials[t];
    for (int t = threadIdx.x; t < n;     t += 256) c += (y_true[t] == 1);
    fs[threadIdx.x] = s;
    cs[threadIdx.x] = c;
    __syncthreads();
    for (int off = 128; off >= 1; off >>= 1) {
        if ((int)threadIdx.x < off) {
            fs[threadIdx.x] += fs[threadIdx.x + off];
            cs[threadIdx.x] += cs[threadIdx.x + off];
        }
        __syncthreads();
    }
    if (threadIdx.x == 0) {
        float npos = (float)cs[0];
        float nneg = (float)(n - cs[0]);
        out[0] = fs[0] / (npos * nneg);
    }
}

// ---------------------------------------------------------------------------
extern "C" void kernel_launch(void* const* d_in, const int* in_sizes, int n_in,
                              void* d_out, int out_size, void* d_ws, size_t ws_size,
                              hipStream_t stream) {
    const float* y_pred = (const float*)d_in[0];
    const int*   y_true = (const int*)d_in[1];
    int n = in_sizes[0];                       // 16384

    int padB = ((n + STRIP - 1) / STRIP) * STRIP;   // multiple of 512 (=> 128, 16)
    float* qpos     = (float*)d_ws;                 // [padB]
    float* qneg     = qpos + padB;                  // [padB]
    float* partials = qneg + padB;                  // [gridX * jStrips]

    int gridX   = padB / ITILE;                     // 128 for B=16384
    int jStrips = padB / STRIP;                     // 32  for B=16384
    int nPart   = gridX * jStrips;                  // 4096

    auc_prep_kernel<<<(padB + 255) / 256, 256, 0, stream>>>(
        y_pred, y_true, qpos, qneg, n, padB);

    dim3 grid(gridX, jStrips);
    auc_pair_kernel<<<grid, 256, 0, stream>>>(qpos, qneg, partials);

    auc_finalize_kernel<<<1, 256, 0, stream>>>(
        y_true, partials, nPart, n, (float*)d_out);
}